// GAT_67851893342523
// MI455X (gfx1250) — compile-verified
//
#include <hip/hip_runtime.h>
#include <hip/hip_bf16.h>

// ---------------------------------------------------------------------------
// GAT forward for MI455X (gfx1250, wave32). GEMMs use V_WMMA_F32_16X16X4_F32
// (f32-precision matrix core path) so results track the f32 JAX reference.
// ---------------------------------------------------------------------------

typedef __attribute__((ext_vector_type(2))) float v2f;
typedef __attribute__((ext_vector_type(8))) float v8f;

static constexpr int N_NODES = 8192;
static constexpr int NNB     = 17;     // neighbors per dst (incl. self loop)
static constexpr int IN_F    = 512;
static constexpr int HID_F   = 256;
static constexpr int OUT_F   = 128;
static constexpr float NEG_SLOPE = 0.2f;

// ---------------------------------------------------------------------------
// 1) Deterministic edge-list extraction from the dense adjacency.
//    One block per row; 32 coalesced rounds of 256 columns; ballot/popcount
//    compaction preserves column order (== jnp.nonzero order).
// ---------------------------------------------------------------------------
__global__ void gat_build_edges(const float* __restrict__ adj,
                                int* __restrict__ edge_src) {
    const int row  = blockIdx.x;
    const int t    = threadIdx.x;        // 256 threads
    const int lane = t & 31;
    const int wv   = t >> 5;             // 8 waves

    __shared__ int waveTot[8];
    __shared__ int running;
    if (t == 0) running = 0;
    __syncthreads();

    const float* arow = adj + (size_t)row * N_NODES;
    int* erow = edge_src + row * NNB;

    for (int r = 0; r < N_NODES / 256; ++r) {
        const int col = r * 256 + t;
        const bool hit = arow[col] != 0.0f;
        const unsigned int b32 = (unsigned int)__ballot((int)hit);
        const int lpos = __popc(b32 & ((1u << lane) - 1u));
        if (lane == 0) waveTot[wv] = __popc(b32);
        __syncthreads();
        int prefix = 0;
        for (int i = 0; i < wv; ++i) prefix += waveTot[i];
        if (hit) {
            const int pos = running + prefix + lpos;
            if (pos < NNB) erow[pos] = col;
        }
        __syncthreads();
        if (t == 0) {
            int tot = 0;
            for (int i = 0; i < 8; ++i) tot += waveTot[i];
            running += tot;
        }
        __syncthreads();
    }
}

// ---------------------------------------------------------------------------
// 2) f32 WMMA GEMM: D[M,NC] = A[M,K] @ B[K,NC] (+ bias). One wave computes one
//    16x16 output tile, accumulating K in steps of 4 via v_wmma_f32_16x16x4_f32.
//    Fragment layout (32-bit operands, wave32):
//      A 16x4: lane%16 = M row; VGPR c holds K = k0 + 2*(lane/16) + c
//      B 4x16: lane%16 = N col; VGPR c holds K = k0 + 2*(lane/16) + c
//      C/D  : VGPR r, lane -> (M = r + 8*(lane/16), N = lane%16)
// ---------------------------------------------------------------------------
template <int K, int NC, bool BIAS>
__global__ void gat_gemm_wmma(const float* __restrict__ A,
                              const float* __restrict__ B,
                              const float* __restrict__ bias,
                              float* __restrict__ D, int M) {
    const int lane = threadIdx.x & 31;
    const int wib  = threadIdx.x >> 5;
    const int tile = blockIdx.x * (blockDim.x >> 5) + wib;

    constexpr int TN = NC / 16;
    const int trow = tile / TN;
    const int tcol = tile % TN;
    if (trow * 16 >= M) return;

    const int m     = lane & 15;          // M-row (for A) / N-col (for B)
    const int khalf = (lane >> 4) << 1;   // 0 or 2

    const float* Abase = A + (size_t)(trow * 16 + m) * K + khalf;
    const float* Bbase = B + (size_t)khalf * NC + tcol * 16 + m;

    v8f acc = {0.f, 0.f, 0.f, 0.f, 0.f, 0.f, 0.f, 0.f};

#pragma unroll 4
    for (int k0 = 0; k0 < K; k0 += 4) {
        v2f a, b;
        a[0] = Abase[k0];
        a[1] = Abase[k0 + 1];
        b[0] = Bbase[(size_t)k0 * NC];
        b[1] = Bbase[(size_t)(k0 + 1) * NC];
        acc = __builtin_amdgcn_wmma_f32_16x16x4_f32(
            /*neg_a=*/false, a, /*neg_b=*/false, b,
            /*c_mod=*/(short)0, acc, /*reuse_a=*/false, /*reuse_b=*/false);
    }

    const int n     = tcol * 16 + m;
    const int rbase = trow * 16 + ((lane >> 4) << 3);
    const float bv  = BIAS ? bias[n] : 0.0f;
#pragma unroll
    for (int r = 0; r < 8; ++r)
        D[(size_t)(rbase + r) * NC + n] = acc[r] + bv;
}

// ---------------------------------------------------------------------------
// 3) Per-node attention dots: es[i] = h_i . a_src ; ed[i] = h_i . a_dst.
//    One wave per node, shfl_xor reduction (wave32).
// ---------------------------------------------------------------------------
template <int C>
__global__ void gat_node_dots(const float* __restrict__ h,
                              const float* __restrict__ a_src,
                              const float* __restrict__ a_dst,
                              float* __restrict__ es, float* __restrict__ ed) {
    const int wave = (blockIdx.x * blockDim.x + threadIdx.x) >> 5;
    const int lane = threadIdx.x & 31;
    if (wave >= N_NODES) return;

    const float* hr = h + (size_t)wave * C;
    float s1 = 0.f, s2 = 0.f;
#pragma unroll
    for (int c = 0; c < C / 32; ++c) {
        const float v = hr[lane + 32 * c];
        s1 += v * a_src[lane + 32 * c];
        s2 += v * a_dst[lane + 32 * c];
    }
#pragma unroll
    for (int o = 16; o > 0; o >>= 1) {
        s1 += __shfl_xor(s1, o, 32);
        s2 += __shfl_xor(s2, o, 32);
    }
    if (lane == 0) { es[wave] = s1; ed[wave] = s2; }
}

// ---------------------------------------------------------------------------
// 4) Edge softmax + weighted scatter-sum + bias. One wave per dst node;
//    lanes 0..16 own the 17 incoming edges.
// ---------------------------------------------------------------------------
__device__ __forceinline__ float leaky_relu(float v) {
    return v >= 0.0f ? v : NEG_SLOPE * v;
}

template <int C>
__global__ void gat_aggregate(const float* __restrict__ h,
                              const float* __restrict__ es,
                              const float* __restrict__ ed,
                              const int* __restrict__ edge_src,
                              const float* __restrict__ bias,
                              float* __restrict__ out) {
    const int wave = (blockIdx.x * blockDim.x + threadIdx.x) >> 5;
    const int lane = threadIdx.x & 31;
    if (wave >= N_NODES) return;

    const int* srcs = edge_src + wave * NNB;
    const int mySrc = (lane < NNB) ? srcs[lane] : 0;

    const float edst = ed[wave];
    float e = (lane < NNB) ? leaky_relu(es[mySrc] + edst) : -3.4e38f;

    // segment max
    float mx = e;
#pragma unroll
    for (int o = 16; o > 0; o >>= 1) mx = fmaxf(mx, __shfl_xor(mx, o, 32));

    const float w = (lane < NNB) ? expf(e - mx) : 0.0f;
    float s = w;
#pragma unroll
    for (int o = 16; o > 0; o >>= 1) s += __shfl_xor(s, o, 32);
    const float coef = w / s;

    float acc[C / 32];
#pragma unroll
    for (int c = 0; c < C / 32; ++c) acc[c] = 0.0f;

#pragma unroll
    for (int j = 0; j < NNB; ++j) {
        const float cj = __shfl(coef, j, 32);
        const int   sj = __shfl(mySrc, j, 32);
        const float* hr = h + (size_t)sj * C;
#pragma unroll
        for (int c = 0; c < C / 32; ++c)
            acc[c] += cj * hr[lane + 32 * c];
    }

    float* orow = out + (size_t)wave * C;
#pragma unroll
    for (int c = 0; c < C / 32; ++c)
        orow[lane + 32 * c] = acc[c] + bias[lane + 32 * c];
}

// ---------------------------------------------------------------------------
// Orchestration
// ---------------------------------------------------------------------------
extern "C" void kernel_launch(void* const* d_in, const int* in_sizes, int n_in,
                              void* d_out, int out_size, void* d_ws, size_t ws_size,
                              hipStream_t stream) {
    const float* x      = (const float*)d_in[0];
    const float* adj    = (const float*)d_in[1];
    const float* W_emb  = (const float*)d_in[2];
    const float* b_emb  = (const float*)d_in[3];
    const float* W_h    = (const float*)d_in[4];
    const float* asrc_h = (const float*)d_in[5];
    const float* adst_h = (const float*)d_in[6];
    const float* b_h    = (const float*)d_in[7];
    const float* W_o    = (const float*)d_in[8];
    const float* asrc_o = (const float*)d_in[9];
    const float* adst_o = (const float*)d_in[10];
    const float* b_o    = (const float*)d_in[11];

    // Workspace layout (bytes):
    //   [0, 1MB)        edge_src  (8192*17 ints = 557056 B)
    //   [1MB, 9MB)      h   (8192*256 f32)
    //   [9MB, 17MB)     hw  (8192*256 f32; reused as 8192*128 for output layer)
    //   [17MB, ...)     es, ed (8192 f32 each)
    char* ws = (char*)d_ws;
    constexpr size_t HBYTES = (size_t)N_NODES * HID_F * sizeof(float); // 8 MB
    int*   edge_src = (int*)(ws);
    float* h        = (float*)(ws + (1u << 20));
    float* hw       = (float*)(ws + (1u << 20) + HBYTES);
    float* es       = (float*)(ws + (1u << 20) + 2 * HBYTES);
    float* ed       = (float*)(ws + (1u << 20) + 2 * HBYTES + 32768);

    const dim3 blk(256);

    // 1) adjacency -> edge lists (coalesced, deterministic)
    gat_build_edges<<<dim3(N_NODES), blk, 0, stream>>>(adj, edge_src);

    // 2) embedding: h = x @ W_emb + b_emb   (8192x512 @ 512x256)
    gat_gemm_wmma<IN_F, HID_F, true>
        <<<dim3((N_NODES / 16) * (HID_F / 16) / 8), blk, 0, stream>>>(
            x, W_emb, b_emb, h, N_NODES);

    // 3) two hidden GAT layers (HID -> HID)
    for (int l = 0; l < 2; ++l) {
        gat_gemm_wmma<HID_F, HID_F, false>
            <<<dim3((N_NODES / 16) * (HID_F / 16) / 8), blk, 0, stream>>>(
                h, W_h + (size_t)l * HID_F * HID_F, nullptr, hw, N_NODES);
        gat_node_dots<HID_F><<<dim3(N_NODES / 8), blk, 0, stream>>>(
            hw, asrc_h + l * HID_F, adst_h + l * HID_F, es, ed);
        gat_aggregate<HID_F><<<dim3(N_NODES / 8), blk, 0, stream>>>(
            hw, es, ed, edge_src, b_h + l * HID_F, h);
    }

    // 4) output GAT layer (HID -> OUT), result straight into d_out
    gat_gemm_wmma<HID_F, OUT_F, false>
        <<<dim3((N_NODES / 16) * (OUT_F / 16) / 8), blk, 0, stream>>>(
            h, W_o, nullptr, hw, N_NODES);
    gat_node_dots<OUT_F><<<dim3(N_NODES / 8), blk, 0, stream>>>(
        hw, asrc_o, adst_o, es, ed);
    gat_aggregate<OUT_F><<<dim3(N_NODES / 8), blk, 0, stream>>>(
        hw, es, ed, edge_src, b_o, (float*)d_out);
}